// Generator_46188078301591
// MI455X (gfx1250) — compile-verified
//
#include <hip/hip_runtime.h>
#include <math.h>

// Problem constants (match reference)
#define BB    32
#define LL    512
#define DD    1024
#define NHH   1024
#define EE    128      // T*(1+C)
#define NCOLC 120      // T*C
#define KTOK  4        // L/E
#define TOPKK 6
#define MROWS (BB*NCOLC)   // 3840

typedef float v2f __attribute__((ext_vector_type(2)));
typedef float v8f __attribute__((ext_vector_type(8)));

// CDNA5 full-precision matrix op: D(16x16,f32) = A(16x4,f32) * B(4x16,f32) + C
__device__ __forceinline__ v8f wmma_f32(v2f a, v2f b, v8f c) {
  return __builtin_amdgcn_wmma_f32_16x16x4_f32(
      /*neg_a=*/false, a, /*neg_b=*/false, b,
      /*c_mod=*/(short)0, c, /*reuse_a=*/false, /*reuse_b=*/false);
}

// ---------------------------------------------------------------------------
// Kernel 1: encoded_entity[b,e,:] = sum_k q_enc[b, tok[e,k], :]
// grid = B*E blocks, 256 threads; one float4 per thread (D=1024).
// ---------------------------------------------------------------------------
__global__ __launch_bounds__(256) void encode_kernel(
    const float* __restrict__ q, const int* __restrict__ tok_idx,
    float* __restrict__ ent) {
  const int be = blockIdx.x;
  const int b  = be >> 7;     // / EE
  const int e  = be & (EE-1);
  const int t0 = tok_idx[e*KTOK+0];
  const int t1 = tok_idx[e*KTOK+1];
  const int t2 = tok_idx[e*KTOK+2];
  const int t3 = tok_idx[e*KTOK+3];
  const int d  = threadIdx.x * 4;
  const size_t base = (size_t)b * LL * DD;
  float4 a0 = *(const float4*)(q + base + (size_t)t0*DD + d);
  float4 a1 = *(const float4*)(q + base + (size_t)t1*DD + d);
  float4 a2 = *(const float4*)(q + base + (size_t)t2*DD + d);
  float4 a3 = *(const float4*)(q + base + (size_t)t3*DD + d);
  float4 s;
  s.x = a0.x + a1.x + a2.x + a3.x;
  s.y = a0.y + a1.y + a2.y + a3.y;
  s.z = a0.z + a1.z + a2.z + a3.z;
  s.w = a0.w + a1.w + a2.w + a3.w;
  *(float4*)(ent + (size_t)be*DD + d) = s;
}

// ---------------------------------------------------------------------------
// Kernel 2: per 16-row M-tile (row = b*120+c):
//   fused = ent[rowC] @ W_col + ent[rowT] @ W_table + b_col + b_table
//   att   = relu(fused @ W_o1 + b_o1) @ W_o2 + b_o2
// 256 threads = 8 waves; wave w owns N slice [w*128, w*128+128).
// ---------------------------------------------------------------------------
#define KP   256            // K panel staged in LDS
#define KPP  260            // padded row stride (260 % 64 == 4 -> no bank conflicts)
#define FP   1028           // fused tile stride  (1028 % 64 == 4)

__global__ __launch_bounds__(256) void att_kernel(
    const float* __restrict__ ent,
    const int*   __restrict__ col_ent_idx, const int* __restrict__ col_tbl_idx,
    const float* __restrict__ W_table, const float* __restrict__ b_table,
    const float* __restrict__ W_col,   const float* __restrict__ b_col,
    const float* __restrict__ W_o1,    const float* __restrict__ b_o1,
    const float* __restrict__ W_o2,    const float* __restrict__ b_o2,
    float* __restrict__ att) {
  __shared__ __align__(16) float sAc[16][KPP];   // gathered A (col entity) panel
  __shared__ __align__(16) float sAt[16][KPP];   // gathered A (table entity) panel
  __shared__ __align__(16) float sF [16][FP];    // fused tile (16 x 1024)
  __shared__ float sAtt[16];
  __shared__ int   sRowC[16], sRowT[16];

  const int tid  = threadIdx.x;
  const int lane = tid & 31;
  const int wave = tid >> 5;
  const int half = lane >> 4;    // K-half for A/B fragments
  const int mrow = lane & 15;    // A: M row, B/C: N column within tile
  const int m0   = blockIdx.x * 16;

  if (tid < 16) {
    int r = m0 + tid;
    int b = r / NCOLC;
    int c = r - b * NCOLC;
    sRowC[tid] = b * EE + col_ent_idx[c];
    sRowT[tid] = b * EE + col_tbl_idx[c];
    sAtt[tid]  = 0.0f;
  }
  __syncthreads();

  const int n_base = wave * 128;

  // ---- Phase 1: fused tile ----
  v8f acc[8];
  #pragma unroll
  for (int nt = 0; nt < 8; ++nt) {
    const int n = n_base + nt*16 + mrow;
    const float bias = b_col[n] + b_table[n];
    #pragma unroll
    for (int v = 0; v < 8; ++v) acc[nt][v] = bias;
  }

  for (int kc = 0; kc < DD; kc += KP) {
    // cooperative stage of both 16x256 A panels (coalesced b128 loads)
    #pragma unroll
    for (int it = 0; it < 4; ++it) {
      const int idx = tid + it*256;       // 0..1023
      const int row = idx >> 6;           // 0..15
      const int c4  = (idx & 63) * 4;     // 0..252
      float4 vc = *(const float4*)(ent + (size_t)sRowC[row]*DD + kc + c4);
      float4 vt = *(const float4*)(ent + (size_t)sRowT[row]*DD + kc + c4);
      *(float4*)&sAc[row][c4] = vc;
      *(float4*)&sAt[row][c4] = vt;
    }
    __syncthreads();

    for (int k = 0; k < KP; k += 4) {
      const v2f ac = *(const v2f*)&sAc[mrow][k + half*2];  // ds_load_b64
      const v2f at = *(const v2f*)&sAt[mrow][k + half*2];
      const int kg = kc + k + half*2;
      #pragma unroll
      for (int nt = 0; nt < 8; ++nt) {
        const int n = n_base + nt*16 + mrow;
        v2f bc, bt;
        bc[0] = W_col  [(size_t)kg     * DD + n];
        bc[1] = W_col  [(size_t)(kg+1) * DD + n];
        bt[0] = W_table[(size_t)kg     * DD + n];
        bt[1] = W_table[(size_t)(kg+1) * DD + n];
        acc[nt] = wmma_f32(ac, bc, acc[nt]);
        acc[nt] = wmma_f32(at, bt, acc[nt]);
      }
    }
    __syncthreads();
  }

  // scatter fused tile into LDS: C-layout VGPR v -> M = v + 8*half, col = N
  #pragma unroll
  for (int nt = 0; nt < 8; ++nt) {
    const int n = n_base + nt*16 + mrow;
    #pragma unroll
    for (int v = 0; v < 8; ++v) sF[v + half*8][n] = acc[nt][v];
  }
  __syncthreads();

  // ---- Phase 2: h = relu(fused @ W_o1 + b_o1); att = h @ W_o2 ----
  v8f acc2[8];
  #pragma unroll
  for (int nt = 0; nt < 8; ++nt) {
    const float bias = b_o1[n_base + nt*16 + mrow];
    #pragma unroll
    for (int v = 0; v < 8; ++v) acc2[nt][v] = bias;
  }

  for (int k = 0; k < DD; k += 4) {
    const v2f af = *(const v2f*)&sF[mrow][k + half*2];
    const int kg = k + half*2;
    #pragma unroll
    for (int nt = 0; nt < 8; ++nt) {
      const int n = n_base + nt*16 + mrow;
      v2f bo;
      bo[0] = W_o1[(size_t)kg     * NHH + n];
      bo[1] = W_o1[(size_t)(kg+1) * NHH + n];
      acc2[nt] = wmma_f32(af, bo, acc2[nt]);
    }
  }

  // epilogue: relu, scale by W_o2[n], reduce over N
  float r[8] = {0,0,0,0,0,0,0,0};
  #pragma unroll
  for (int nt = 0; nt < 8; ++nt) {
    const float w2 = W_o2[n_base + nt*16 + mrow];
    #pragma unroll
    for (int v = 0; v < 8; ++v) {
      float h = acc2[nt][v];
      h = h > 0.0f ? h : 0.0f;
      r[v] += h * w2;
    }
  }
  // reduce across the 16 lanes of each half (xor masks stay within halves)
  #pragma unroll
  for (int v = 0; v < 8; ++v) {
    float x = r[v];
    x += __shfl_xor(x, 8, 32);
    x += __shfl_xor(x, 4, 32);
    x += __shfl_xor(x, 2, 32);
    x += __shfl_xor(x, 1, 32);
    if (mrow == 0) atomicAdd(&sAtt[v + half*8], x);  // ds_add_f32 across waves/halves
  }
  __syncthreads();
  if (tid < 16) att[m0 + tid] = sAtt[tid] + b_o2[0];
}

// ---------------------------------------------------------------------------
// Kernel 3: topk = sigmoid(relu(cls @ W_t1 + b_t1) @ W_t2 + b_t2)  (tiny)
// grid = B blocks of 256 threads.
// ---------------------------------------------------------------------------
__global__ __launch_bounds__(256) void topk_kernel(
    const float* __restrict__ q,
    const float* __restrict__ W_t1, const float* __restrict__ b_t1,
    const float* __restrict__ W_t2, const float* __restrict__ b_t2,
    float* __restrict__ out_topk) {
  __shared__ __align__(16) float s_cls[DD];
  __shared__ float s_part[8][TOPKK];
  const int b   = blockIdx.x;
  const int tid = threadIdx.x;
  {
    const int d = tid * 4;
    *(float4*)&s_cls[d] = *(const float4*)(q + (size_t)b*LL*DD + d);
  }
  __syncthreads();

  float racc[TOPKK] = {0,0,0,0,0,0};
  for (int n = tid; n < NHH; n += 256) {
    float h = b_t1[n];
    for (int k = 0; k < DD; ++k) h = fmaf(s_cls[k], W_t1[(size_t)k*NHH + n], h);
    h = h > 0.0f ? h : 0.0f;
    #pragma unroll
    for (int t = 0; t < TOPKK; ++t) racc[t] += h * W_t2[(size_t)n*TOPKK + t];
  }
  const int lane = tid & 31, wave = tid >> 5;
  #pragma unroll
  for (int t = 0; t < TOPKK; ++t) {
    float x = racc[t];
    x += __shfl_xor(x, 16, 32);
    x += __shfl_xor(x,  8, 32);
    x += __shfl_xor(x,  4, 32);
    x += __shfl_xor(x,  2, 32);
    x += __shfl_xor(x,  1, 32);
    if (lane == 0) s_part[wave][t] = x;
  }
  __syncthreads();
  if (tid < TOPKK) {
    float s = b_t2[tid];
    #pragma unroll
    for (int w = 0; w < 8; ++w) s += s_part[w][tid];
    out_topk[b*TOPKK + tid] = 1.0f / (1.0f + __expf(-s));
  }
}

// ---------------------------------------------------------------------------
extern "C" void kernel_launch(void* const* d_in, const int* in_sizes, int n_in,
                              void* d_out, int out_size, void* d_ws, size_t ws_size,
                              hipStream_t stream) {
  const float* q_enc   = (const float*)d_in[0];
  const int*   tok_idx = (const int*)  d_in[1];
  const int*   col_ent = (const int*)  d_in[2];
  const int*   col_tbl = (const int*)  d_in[3];
  const float* W_table = (const float*)d_in[4];
  const float* b_table = (const float*)d_in[5];
  const float* W_col   = (const float*)d_in[6];
  const float* b_col   = (const float*)d_in[7];
  const float* W_o1    = (const float*)d_in[8];
  const float* b_o1    = (const float*)d_in[9];
  const float* W_o2    = (const float*)d_in[10];
  const float* b_o2    = (const float*)d_in[11];
  const float* W_t1    = (const float*)d_in[12];
  const float* b_t1    = (const float*)d_in[13];
  const float* W_t2    = (const float*)d_in[14];
  const float* b_t2    = (const float*)d_in[15];

  float* att  = (float*)d_out;            // B*NCOL
  float* topk = att + BB*NCOLC;           // B*TOPK
  float* ent  = (float*)d_ws;             // B*E*D scratch (16 MB)

  encode_kernel<<<BB*EE, 256, 0, stream>>>(q_enc, tok_idx, ent);
  att_kernel<<<MROWS/16, 256, 0, stream>>>(ent, col_ent, col_tbl,
                                           W_table, b_table, W_col, b_col,
                                           W_o1, b_o1, W_o2, b_o2, att);
  topk_kernel<<<BB, 256, 0, stream>>>(q_enc, W_t1, b_t1, W_t2, b_t2, topk);
}